// RegularGridInterpolator_62216896250095
// MI455X (gfx1250) — compile-verified
//
#include <hip/hip_runtime.h>

// ---------------------------------------------------------------------------
// Trilinear regular-grid interpolation on MI455X (gfx1250, wave32).
//
// One wave32 per point. The per-point blend
//     out[c] = sum_{8 corners} w_corner * grid[c][corner]
// is computed with V_WMMA_F32_16X16X4_F32:
//   A (16x4)  : M = 16 channels, K = 4 corners (VGPR0=K even, VGPR1=K odd;
//               lane-half selects K{0,1} vs K{2,3}) -> one b64 gather per
//               lane supplies a whole A tile (k,k+1 are memory-adjacent).
//   B (4x16)  : per-corner weights broadcast along N (all columns equal).
//   Two chained WMMAs cover the 8 corners (i_lo, i_hi); two C accumulators
//   cover channels 0-15 and 16-31.
//
// Memory: 4 wave-wide b64 gathers/point = 1024 useful bytes = minimum;
// working set (upper octant, 33.5 MB) is L2-resident (192 MB L2).
// This revision is branch-free (v_cndmask instead of exec-mask control flow)
// and uses 32-bit offsets so gathers/stores use SGPR-base + VGPR32-offset
// (GVS) addressing — trims the VALU/SALU overlay down toward the VMEM floor.
// ---------------------------------------------------------------------------

typedef __attribute__((ext_vector_type(2))) float v2f;
typedef __attribute__((ext_vector_type(4))) float v4f;
typedef __attribute__((ext_vector_type(8))) float v8f;
// 4-byte-aligned float2 view (k index may be odd): unaligned b64 load
typedef float v2fu __attribute__((ext_vector_type(2), aligned(4)));

__device__ __forceinline__ float tickf(int i) {
    return (float)i * 0.015625f - 1.0f;   // exact in f32
}

// searchsorted(ticks, x, side='left') with clipping, per reference.
// Branch-free: all fix-ups lower to v_cmp + v_cndmask.
__device__ __forceinline__ void bucket1d(float x, int& il, float& dl, float& dr) {
    float t = (x + 1.0f) * 64.0f;
    int ir = (int)ceilf(t);
    ir = ir < 0 ? 0 : ir;
    ir = ir > 127 ? 127 : ir;
    // exact fix-up: want smallest i with tick(i) >= x (clipped to 127)
    int dec = (ir > 0) & (tickf(ir - 1) >= x);
    ir -= dec;
    int inc = (ir < 127) & (tickf(ir) < x);
    ir += inc;
    il = ir - 1;
    il = il < 0 ? 0 : il;
    dl = fmaxf(x - tickf(il), 0.0f);
    dr = fmaxf(tickf(ir) - x, 0.0f);
    int bz = (dl == 0.0f) & (dr == 0.0f);   // exact-tick degenerate case
    dl = bz ? 1.0f : dl;
    dr = bz ? 1.0f : dr;
}

__global__ __launch_bounds__(256)
void RegularGridInterpolator_62216896250095_kernel(const float* __restrict__ pts,
                                                   const float* __restrict__ grid,
                                                   float* __restrict__ out,
                                                   int N)
{
    const int wave = (int)(threadIdx.x >> 5);
    const int lane = (int)(threadIdx.x & 31);
    const int p    = (int)blockIdx.x * 8 + wave;
    if (p >= N) return;                       // wave-uniform: EXEC stays all-1s

    const float x = pts[p * 3 + 0];
    const float y = pts[p * 3 + 1];
    const float z = pts[p * 3 + 2];

    int il, jl, kl;
    float dlx, drx, dly, dry, dlz, drz;
    bucket1d(x, il, dlx, drx);
    bucket1d(y, jl, dly, dry);
    bucket1d(z, kl, dlz, drz);

    // Fold the final normalization into the corner weights (IEEE f32 divide,
    // matching the reference's division semantics).
    const float scale = 1.0f / ((dlx + drx) * (dly + dry) * (dlz + drz));
    const float fx0 = drx * scale;            // i = il   (i_on=0 -> dist_right)
    const float fx1 = dlx * scale;            // i = il+1 (i_on=1 -> dist_left)

    // B-matrix (weights), broadcast along N. Lane-half picks the j corner,
    // vector component picks the k corner (matches A's K placement).
    const int   jsel = lane >> 4;             // 0: j_lo (K0,K1)  1: j_hi (K2,K3)
    const float wj   = jsel ? dly : dry;
    v2f b0, b1;
    b0.x = fx0 * wj * drz;  b0.y = fx0 * wj * dlz;   // i = il
    b1.x = fx1 * wj * drz;  b1.y = fx1 * wj * dlz;   // i = il+1

    // A-matrix gathers: grid[ch][i][j][k], strides ch:128^3, i:128^2, j:128.
    // 32-bit offsets (max 64M floats) -> GVS addressing (SGPR base + VGPR32).
    const int offBase = il * 16384 + (jl + jsel) * 128 + kl;
    const int offLo   = offBase + (lane & 15) * 2097152;   // channels 0-15
    const int offHi   = offLo + 16 * 2097152;              // channels 16-31

    v2f a00 = *(const v2fu*)(grid + offLo);            // (il,   ch 0-15)
    v2f a01 = *(const v2fu*)(grid + offLo + 16384);    // (il+1, ch 0-15)
    v2f a10 = *(const v2fu*)(grid + offHi);            // (il,   ch 16-31)
    v2f a11 = *(const v2fu*)(grid + offHi + 16384);    // (il+1, ch 16-31)

    v8f c0 = {};
    v8f c1 = {};
    // D = A*B + C ; chain the two i-planes through the accumulator.
    c0 = __builtin_amdgcn_wmma_f32_16x16x4_f32(false, a00, false, b0, (short)0, c0, false, false);
    c0 = __builtin_amdgcn_wmma_f32_16x16x4_f32(false, a01, false, b1, (short)0, c0, false, false);
    c1 = __builtin_amdgcn_wmma_f32_16x16x4_f32(false, a10, false, b0, (short)0, c1, false, false);
    c1 = __builtin_amdgcn_wmma_f32_16x16x4_f32(false, a11, false, b1, (short)0, c1, false, false);

    // D layout: lane L holds column N=L%16; VGPR r = row M=r (+8 for lanes>=16).
    // All columns identical -> lanes 0 and 16 hold the full 16-channel result
    // of each accumulator in their 8 D VGPRs.
    if ((lane & 15) == 0) {
        float* o = out + p * 32 + (lane >> 4) * 8;     // ch base 0 / 8
        v4f r0 = { c0[0], c0[1], c0[2], c0[3] };
        v4f r1 = { c0[4], c0[5], c0[6], c0[7] };
        v4f r2 = { c1[0], c1[1], c1[2], c1[3] };
        v4f r3 = { c1[4], c1[5], c1[6], c1[7] };
        *(v4f*)(o + 0)  = r0;   // channels base+0  .. base+3
        *(v4f*)(o + 4)  = r1;   // channels base+4  .. base+7
        *(v4f*)(o + 16) = r2;   // channels base+16 .. base+19
        *(v4f*)(o + 20) = r3;   // channels base+20 .. base+23
    }
}

extern "C" void kernel_launch(void* const* d_in, const int* in_sizes, int n_in,
                              void* d_out, int out_size, void* d_ws, size_t ws_size,
                              hipStream_t stream) {
    const float* pts  = (const float*)d_in[0];   // [N,3] f32
    const float* grid = (const float*)d_in[1];   // [32,128,128,128] f32
    float* out        = (float*)d_out;           // [N,32] f32

    const int N = in_sizes[0] / 3;
    const int blocks = (N + 7) / 8;              // 8 waves (points) per 256-thread block
    hipLaunchKernelGGL(RegularGridInterpolator_62216896250095_kernel,
                       dim3(blocks), dim3(256), 0, stream,
                       pts, grid, out, N);
}